// TokenGCN_90683939487935
// MI455X (gfx1250) — compile-verified
//
#include <hip/hip_runtime.h>
#include <hip/hip_bf16.h>

typedef __attribute__((ext_vector_type(2))) float v2f;
typedef __attribute__((ext_vector_type(8))) float v8f;

#define B_   64
#define N_   512
#define C_   768
#define H_   512
#define O_   768
#define NSPLIT 8
#define NCHUNK (N_ / NSPLIT)   // 64

// ---------------------------------------------------------------------------
// Kernel 1: partial mean over nodes.  grid = (1, B, NSPLIT), block = 192.
// Each thread owns 4 consecutive channels (float4), sums NCHUNK nodes.
// partials layout: [NSPLIT][B][C] floats.
// ---------------------------------------------------------------------------
__global__ void __launch_bounds__(192) k_reduce_partial(const float* __restrict__ x,
                                                        float* __restrict__ partials) {
    const int c4   = threadIdx.x * 4;          // 0..764
    const int b    = blockIdx.y;
    const int s    = blockIdx.z;
    const float* xp = x + (size_t)b * N_ * C_ + (size_t)(s * NCHUNK) * C_ + c4;
    float4 acc = make_float4(0.f, 0.f, 0.f, 0.f);
    #pragma unroll 4
    for (int n = 0; n < NCHUNK; ++n) {
        float4 v = *(const float4*)(xp + (size_t)n * C_);
        acc.x += v.x; acc.y += v.y; acc.z += v.z; acc.w += v.w;
    }
    *(float4*)(partials + ((size_t)s * B_ + b) * C_ + c4) = acc;
}

// ---------------------------------------------------------------------------
// Kernel 2: fold NSPLIT partials, scale by 1/N.  xbar layout: [B][C].
// ---------------------------------------------------------------------------
__global__ void __launch_bounds__(256) k_reduce_final(const float* __restrict__ partials,
                                                      float* __restrict__ xbar) {
    const int idx = blockIdx.x * blockDim.x + threadIdx.x;   // over B*C/4
    const int c4  = idx * 4;
    if (c4 >= B_ * C_) return;
    float4 acc = make_float4(0.f, 0.f, 0.f, 0.f);
    #pragma unroll
    for (int s = 0; s < NSPLIT; ++s) {
        float4 v = *(const float4*)(partials + (size_t)s * B_ * C_ + c4);
        acc.x += v.x; acc.y += v.y; acc.z += v.z; acc.w += v.w;
    }
    const float inv = 1.0f / (float)N_;
    acc.x *= inv; acc.y *= inv; acc.z *= inv; acc.w *= inv;
    *(float4*)(xbar + c4) = acc;
}

// ---------------------------------------------------------------------------
// Kernel 3: tiny GEMM  Out(64 x Nout) = relu(A(64 x K) * W(K x Nout) + bias)
// One wave (32 threads) per 16x16 output tile, V_WMMA_F32_16X16X4_F32.
// grid = (4, Nout/16), block = 32.  EXEC stays all-ones (no divergence).
// ---------------------------------------------------------------------------
__global__ void __launch_bounds__(32) k_gemm_relu(const float* __restrict__ A,
                                                  const float* __restrict__ W,
                                                  const float* __restrict__ bias,
                                                  float* __restrict__ Out,
                                                  int K, int Nout) {
    const int lane = threadIdx.x;
    const int half = lane >> 4;          // 0: lanes 0-15, 1: lanes 16-31
    const int l    = lane & 15;
    const int col  = blockIdx.y * 16 + l;
    const float* Arow = A + (size_t)(blockIdx.x * 16 + l) * K;

    v8f acc = {};
    for (int k = 0; k < K; k += 4) {
        const int ka = k + half * 2;
        // A-frag: 16x4, row M = l, K = ka, ka+1   (8B-aligned: ka even)
        v2f a = *(const v2f*)(Arow + ka);
        // B-frag: 4x16, rows K = ka, ka+1 at column N = col
        v2f b;
        b.x = W[(size_t)ka * Nout + col];
        b.y = W[(size_t)(ka + 1) * Nout + col];
        acc = __builtin_amdgcn_wmma_f32_16x16x4_f32(
            /*neg_a=*/false, a, /*neg_b=*/false, b,
            /*c_mod=*/(short)0, acc, /*reuse_a=*/false, /*reuse_b=*/false);
    }

    const float bv = bias[col];
    #pragma unroll
    for (int v = 0; v < 8; ++v) {
        const int m = blockIdx.x * 16 + v + half * 8;   // C/D VGPR layout
        float r = acc[v] + bv;
        r = fmaxf(r, 0.0f);
        Out[(size_t)m * Nout + col] = r;
    }
}

// ---------------------------------------------------------------------------
// Kernel 4: broadcast h3 (B x O) to out (B x 128 x O), float4-vectorized.
// ---------------------------------------------------------------------------
__global__ void __launch_bounds__(256) k_broadcast(const float* __restrict__ h3,
                                                   float* __restrict__ out) {
    const int idx = blockIdx.x * blockDim.x + threadIdx.x;   // over B*128*O/4
    const int total = B_ * 128 * (O_ / 4);
    if (idx >= total) return;
    const int o4 = idx % (O_ / 4);
    const int bi = idx / (O_ / 4);                           // b*128 + i
    const int b  = bi / 128;
    float4 v = *(const float4*)(h3 + (size_t)b * O_ + o4 * 4);
    *(float4*)(out + (size_t)idx * 4) = v;
}

// ---------------------------------------------------------------------------
extern "C" void kernel_launch(void* const* d_in, const int* in_sizes, int n_in,
                              void* d_out, int out_size, void* d_ws, size_t ws_size,
                              hipStream_t stream) {
    const float* x  = (const float*)d_in[0];
    const float* W1 = (const float*)d_in[1];
    const float* b1 = (const float*)d_in[2];
    const float* W2 = (const float*)d_in[3];
    const float* b2 = (const float*)d_in[4];
    const float* W3 = (const float*)d_in[5];
    const float* b3 = (const float*)d_in[6];
    float* out = (float*)d_out;

    // Workspace layout (floats)
    float* partials = (float*)d_ws;                       // NSPLIT*B*C = 393216
    float* xbar = partials + (size_t)NSPLIT * B_ * C_;    // B*C  = 49152
    float* h1   = xbar + (size_t)B_ * C_;                 // B*H  = 32768
    float* h2   = h1 + (size_t)B_ * H_;                   // B*H  = 32768
    float* h3   = h2 + (size_t)B_ * H_;                   // B*O  = 49152

    // 1) mean over nodes (two-phase, deterministic)
    dim3 g1(1, B_, NSPLIT);
    k_reduce_partial<<<g1, 192, 0, stream>>>(x, partials);
    int n2 = (B_ * C_ / 4 + 255) / 256;
    k_reduce_final<<<n2, 256, 0, stream>>>(partials, xbar);

    // 2) MLP chain with fp32 WMMA
    k_gemm_relu<<<dim3(B_ / 16, H_ / 16), 32, 0, stream>>>(xbar, W1, b1, h1, C_, H_);
    k_gemm_relu<<<dim3(B_ / 16, H_ / 16), 32, 0, stream>>>(h1,   W2, b2, h2, H_, H_);
    k_gemm_relu<<<dim3(B_ / 16, O_ / 16), 32, 0, stream>>>(h2,   W3, b3, h3, H_, O_);

    // 3) broadcast to (B, 128, O)
    int n4 = (B_ * 128 * (O_ / 4) + 255) / 256;
    k_broadcast<<<n4, 256, 0, stream>>>(h3, out);
}